// ElasticGNN_56427280335488
// MI455X (gfx1250) — compile-verified
//
#include <hip/hip_runtime.h>

typedef float v2f __attribute__((ext_vector_type(2)));
typedef float v8f __attribute__((ext_vector_type(8)));

#define IN_FEATS 512
#define HID      256
#define FEAT     64            // OUT_FEATS
#define GAMMA    0.25f         // 1/(1+LAMBDA2)
#define BETA     2.0f          // 1/(2*GAMMA)
#define LAM1     3.0f
#define LAM2     3.0f

// ---------------- elementwise helpers ----------------
__global__ void fill_f32(float* __restrict__ p, float v, int n) {
  int i = blockIdx.x * 256 + threadIdx.x;
  if (i < n) p[i] = v;
}

// Wt[n*K + k] = W[k*N + n]
__global__ void transpose_f32(const float* __restrict__ W, float* __restrict__ Wt, int K, int N) {
  int i = blockIdx.x * 256 + threadIdx.x;
  if (i < K * N) { int k = i / N, n = i - k * N; Wt[(size_t)n * K + k] = W[i]; }
}

__global__ void degree_count(const int* __restrict__ row, const int* __restrict__ col,
                             float* __restrict__ deg, int E) {
  int e = blockIdx.x * 256 + threadIdx.x;
  if (e < E) { atomicAdd(&deg[row[e]], 1.0f); atomicAdd(&deg[col[e]], 1.0f); }
}

__global__ void rsqrt_inplace(float* __restrict__ d, int n) {
  int i = blockIdx.x * 256 + threadIdx.x;
  if (i < n) d[i] = rsqrtf(d[i]);
}

// y = (1-GAMMA)*xk + GAMMA*hh - GAMMA*LAM2*t     (t = inc_t_mv(inc_mv(xk)))
__global__ void y_update(const float* __restrict__ xk, const float* __restrict__ hh,
                         const float* __restrict__ t, float* __restrict__ y, int n) {
  int i = blockIdx.x * 256 + threadIdx.x;
  if (i < n) y[i] = (1.0f - GAMMA) * xk[i] + GAMMA * hh[i] - (GAMMA * LAM2) * t[i];
}

// out = y - GAMMA*t                              (t = inc_t_mv(z))
__global__ void x_update(const float* __restrict__ y, const float* __restrict__ t,
                         float* __restrict__ out, int n) {
  int i = blockIdx.x * 256 + threadIdx.x;
  if (i < n) out[i] = y[i] - GAMMA * t[i];
}

// ---------------- f32 WMMA GEMM: C[M,ldc] = A[M,KDIM] * Bt[N,KDIM]^T + bias ----------------
// One wave per 16x16 output tile, V_WMMA_F32_16X16X4_F32, K stepped by 4.
// A-frag (16x4 f32): lanes 0-15 hold M=l, K=0,1 ; lanes 16-31 hold M=l, K=2,3.
// B-frag (4x16 f32): lanes 0-15 hold N=l, K=0,1 ; lanes 16-31 hold N=l, K=2,3. (Bt is N-major)
// D (16x16 f32): VGPR r, lanes 0-15 -> (M=r, N=l); lanes 16-31 -> (M=r+8, N=l).
template<int KDIM, bool RELU, bool DUP>
__global__ __launch_bounds__(32)
void gemm16_wmma(const float* __restrict__ A, const float* __restrict__ Bt,
                 const float* __restrict__ bias, float* __restrict__ C,
                 float* __restrict__ C2, int ldc) {
  const int lane = threadIdx.x & 31;
  const int l    = lane & 15;
  const int half = lane >> 4;
  const size_t m0 = (size_t)blockIdx.x * 16;
  const int    n0 = blockIdx.y * 16;
  const float* ap = A  + (m0 + l) * KDIM + 2 * half;
  const float* bp = Bt + ((size_t)(n0 + l)) * KDIM + 2 * half;
  v8f acc = {0.f, 0.f, 0.f, 0.f, 0.f, 0.f, 0.f, 0.f};
#pragma unroll 8
  for (int k = 0; k < KDIM; k += 4) {
    v2f a = *(const v2f*)(ap + k);
    v2f b = *(const v2f*)(bp + k);
#if defined(__HIP_DEVICE_COMPILE__)
    acc = __builtin_amdgcn_wmma_f32_16x16x4_f32(false, a, false, b, (short)0, acc,
                                                false, false);
#else
    (void)a; (void)b;
#endif
  }
  const float  bv = bias[n0 + l];
  const size_t mb = m0 + 8 * half;
#pragma unroll
  for (int r = 0; r < 8; ++r) {
    float v = acc[r] + bv;
    if (RELU) v = fmaxf(v, 0.0f);
    size_t off = (mb + r) * (size_t)ldc + (size_t)(n0 + l);
    C[off] = v;
    if (DUP) C2[off] = v;
  }
}

// ---------------- EMP edge kernels (one wave32 per edge, 2 feats/lane) ----------------
// t += inc_t_mv(inc_mv(xk)) : m = dr*xk[r] - dc*xk[c]; t[r] += dr*m; t[c] -= dc*m
__global__ void edge_laplacian(const float* __restrict__ xk, const float* __restrict__ dvec,
                               const int* __restrict__ row, const int* __restrict__ col,
                               float* __restrict__ t, int E) {
  int e = blockIdx.x * 8 + (threadIdx.x >> 5);
  int lane = threadIdx.x & 31;
  if (e >= E) return;
  int r = row[e], c = col[e];
  float dr = dvec[r], dc = dvec[c];
  const float2 xr = *(const float2*)(xk + (size_t)r * FEAT + lane * 2);
  const float2 xc = *(const float2*)(xk + (size_t)c * FEAT + lane * 2);
  float mx = dr * xr.x - dc * xc.x;
  float my = dr * xr.y - dc * xc.y;
  float* tr = t + (size_t)r * FEAT + lane * 2;
  float* tc = t + (size_t)c * FEAT + lane * 2;
  atomicAdd(tr,     dr * mx);  atomicAdd(tr + 1,  dr * my);
  atomicAdd(tc,    -dc * mx);  atomicAdd(tc + 1, -dc * my);
}

// z_bar = z + BETA*inc_mv(y); L21-project rows to ball radius LAM1; z = proj;
// then scatter t += inc_t_mv(z) in the same pass (z touched exactly once).
__global__ void edge_z_project(const float* __restrict__ y, float* __restrict__ z,
                               const float* __restrict__ dvec,
                               const int* __restrict__ row, const int* __restrict__ col,
                               float* __restrict__ t, int E, int use_z) {
  int e = blockIdx.x * 8 + (threadIdx.x >> 5);
  int lane = threadIdx.x & 31;
  if (e >= E) return;
  int r = row[e], c = col[e];
  float dr = dvec[r], dc = dvec[c];
  const float2 yr = *(const float2*)(y + (size_t)r * FEAT + lane * 2);
  const float2 yc = *(const float2*)(y + (size_t)c * FEAT + lane * 2);
  float zx = BETA * (dr * yr.x - dc * yc.x);
  float zy = BETA * (dr * yr.y - dc * yc.y);
  float* zp = z + (size_t)e * FEAT + lane * 2;
  if (use_z) { float2 zo = *(const float2*)zp; zx += zo.x; zy += zo.y; }
  // wave-wide row norm over the 64 features
  float s = zx * zx + zy * zy;
  #pragma unroll
  for (int off = 16; off > 0; off >>= 1) s += __shfl_xor(s, off, 32);
  float rn = sqrtf(s);
  float scale = (rn > LAM1) ? (LAM1 / rn) : 1.0f;  // == min(rn,LAM1)/rn (rn==0 -> z stays 0)
  zx *= scale; zy *= scale;
  float2 zn; zn.x = zx; zn.y = zy;
  *(float2*)zp = zn;
  float* tr = t + (size_t)r * FEAT + lane * 2;
  float* tc = t + (size_t)c * FEAT + lane * 2;
  atomicAdd(tr,     dr * zx);  atomicAdd(tr + 1,  dr * zy);
  atomicAdd(tc,    -dc * zx);  atomicAdd(tc + 1, -dc * zy);
}

// ---------------- driver ----------------
extern "C" void kernel_launch(void* const* d_in, const int* in_sizes, int n_in,
                              void* d_out, int out_size, void* d_ws, size_t ws_size,
                              hipStream_t stream) {
  const float* x  = (const float*)d_in[0];
  const float* W1 = (const float*)d_in[1];
  const float* b1 = (const float*)d_in[2];
  const float* W2 = (const float*)d_in[3];
  const float* b2 = (const float*)d_in[4];
  const int*   ei = (const int*)d_in[5];
  const int N = in_sizes[0] / IN_FEATS;     // 100000 (multiple of 16)
  const int E = in_sizes[5] / 2;            // 1600000
  const int* row = ei;
  const int* col = ei + E;

  // workspace carve (256B-aligned); big region is h (N*HID) then reused as z (E*FEAT)
  char* p = (char*)d_ws;
  auto carve = [&](size_t bytes) -> float* {
    float* q = (float*)p; p += (bytes + 255) & ~(size_t)255; return q;
  };
  size_t hB = (size_t)N * HID * 4, zB = (size_t)E * FEAT * 4;
  float* big = carve(hB > zB ? hB : zB);    // h / z
  float* h   = big;
  float* z   = big;
  float* hh  = carve((size_t)N * FEAT * 4);
  float* xk  = carve((size_t)N * FEAT * 4);
  float* yv  = carve((size_t)N * FEAT * 4);
  float* t   = carve((size_t)N * FEAT * 4);
  float* dv  = carve((size_t)N * 4);
  float* W1t = carve((size_t)IN_FEATS * HID * 4);
  float* W2t = carve((size_t)HID * FEAT * 4);
  (void)ws_size; (void)n_in; (void)out_size;

  const int nodeElems = N * FEAT;
  dim3 blk256(256);
  dim3 gNode((nodeElems + 255) / 256);
  dim3 gEdgeWave((E + 7) / 8);              // 8 waves (edges) per 256-thread block

  // weight transposes (tiny)
  transpose_f32<<<(IN_FEATS * HID + 255) / 256, blk256, 0, stream>>>(W1, W1t, IN_FEATS, HID);
  transpose_f32<<<(HID * FEAT + 255) / 256, blk256, 0, stream>>>(W2, W2t, HID, FEAT);

  // degree -> d = (1 + deg_row + deg_col)^-1/2
  fill_f32<<<(N + 255) / 256, blk256, 0, stream>>>(dv, 1.0f, N);
  degree_count<<<(E + 255) / 256, blk256, 0, stream>>>(row, col, dv, E);
  rsqrt_inplace<<<(N + 255) / 256, blk256, 0, stream>>>(dv, N);

  // encoder: h = relu(x@W1+b1); hh = h@W2+b2 (dup into xk = initial state)
  gemm16_wmma<IN_FEATS, true, false><<<dim3(N / 16, HID / 16), dim3(32), 0, stream>>>(
      x, W1t, b1, h, nullptr, HID);
  gemm16_wmma<HID, false, true><<<dim3(N / 16, FEAT / 16), dim3(32), 0, stream>>>(
      h, W2t, b2, hh, xk, FEAT);

  // EMP: K=3 iterations (z implicitly zero in iteration 0 -> use_z=0, no init needed)
  for (int it = 0; it < 3; ++it) {
    fill_f32<<<gNode, blk256, 0, stream>>>(t, 0.0f, nodeElems);
    edge_laplacian<<<gEdgeWave, blk256, 0, stream>>>(xk, dv, row, col, t, E);
    y_update<<<gNode, blk256, 0, stream>>>(xk, hh, t, yv, nodeElems);

    fill_f32<<<gNode, blk256, 0, stream>>>(t, 0.0f, nodeElems);
    edge_z_project<<<gEdgeWave, blk256, 0, stream>>>(yv, z, dv, row, col, t, E,
                                                     it > 0 ? 1 : 0);
    float* outp = (it == 2) ? (float*)d_out : xk;
    x_update<<<gNode, blk256, 0, stream>>>(yv, t, outp, nodeElems);
  }
}